// MLM_9088150798516
// MI455X (gfx1250) — compile-verified
//
#include <hip/hip_runtime.h>
#include <hip/hip_bf16.h>

typedef float v2f __attribute__((ext_vector_type(2)));
typedef float v8f __attribute__((ext_vector_type(8)));

#define NN 80000
#define EE 1280000
#define GG 512
#define DD 64
#define BN_EPS 1e-5f

// ---------------------------------------------------------------------------
// zero fill
// ---------------------------------------------------------------------------
__global__ void gnn_zero_kernel(float* __restrict__ p, int n) {
    int i = blockIdx.x * blockDim.x + threadIdx.x;
    if (i < n) p[i] = 0.0f;
}

// ---------------------------------------------------------------------------
// Layer 0 (GINE): e = edge_attr @ We ; msg = relu(x[src] + e + be)
// atomicAdd into agg[dst].   256 thr = 8 waves, 16 edges/wave, 128 edges/block
// ---------------------------------------------------------------------------
__global__ __launch_bounds__(256)
void gine_edge_kernel(const float* __restrict__ x,
                      const int*   __restrict__ ei,
                      const float* __restrict__ ea,
                      const float* __restrict__ We,
                      const float* __restrict__ be,
                      float*       __restrict__ agg)
{
    __shared__ float sWe[64 * 64];
    __shared__ float sA[8][16 * 65];   // padded edge tiles, one per wave
    __shared__ float sbe[64];

    const int tid  = threadIdx.x;
    const int lane = tid & 31;
    const int wave = tid >> 5;
    const int half = lane >> 4;        // 0..1
    const int l16  = lane & 15;        // 0..15

    for (int i = tid; i < 64 * 64; i += 256) sWe[i] = We[i];
    if (tid < 64) sbe[tid] = be[tid];

    const int ebase = (blockIdx.x * 8 + wave) * 16;

    // stage 16x64 edge_attr tile (coalesced float4)
    for (int it = lane; it < 16 * 16; it += 32) {
        int row = it >> 4, c4 = it & 15;
        float4 v = *(const float4*)(ea + (size_t)(ebase + row) * DD + c4 * 4);
        float* d = &sA[wave][row * 65 + c4 * 4];
        d[0] = v.x; d[1] = v.y; d[2] = v.z; d[3] = v.w;
    }
    __syncthreads();

    v8f acc0 = {}, acc1 = {}, acc2 = {}, acc3 = {};
    const float* Arow = &sA[wave][l16 * 65];
#pragma unroll
    for (int kk = 0; kk < 16; ++kk) {
        const int kc = kk * 4 + half * 2;
        v2f a; a.x = Arow[kc]; a.y = Arow[kc + 1];
        const float* Wr0 = &sWe[kc * 64 + l16];
        const float* Wr1 = Wr0 + 64;
        v2f b0, b1, b2, b3;
        b0.x = Wr0[0];  b0.y = Wr1[0];
        b1.x = Wr0[16]; b1.y = Wr1[16];
        b2.x = Wr0[32]; b2.y = Wr1[32];
        b3.x = Wr0[48]; b3.y = Wr1[48];
        acc0 = __builtin_amdgcn_wmma_f32_16x16x4_f32(false, a, false, b0, (short)0, acc0, false, false);
        acc1 = __builtin_amdgcn_wmma_f32_16x16x4_f32(false, a, false, b1, (short)0, acc1, false, false);
        acc2 = __builtin_amdgcn_wmma_f32_16x16x4_f32(false, a, false, b2, (short)0, acc2, false, false);
        acc3 = __builtin_amdgcn_wmma_f32_16x16x4_f32(false, a, false, b3, (short)0, acc3, false, false);
    }

    // epilogue: +be, gather x[src], relu, atomic scatter into agg[dst]
#pragma unroll
    for (int r = 0; r < 8; ++r) {
        const int erow = r + half * 8;
        const int eid  = ebase + erow;
        const int src  = ei[eid];
        const int dst  = ei[EE + eid];
        const float* xs = x   + (size_t)src * DD;
        float*       ag = agg + (size_t)dst * DD;
        float m;
        m = acc0[r] + sbe[l16]      + xs[l16];      m = m > 0.f ? m : 0.f; atomicAdd(ag + l16,      m);
        m = acc1[r] + sbe[l16 + 16] + xs[l16 + 16]; m = m > 0.f ? m : 0.f; atomicAdd(ag + l16 + 16, m);
        m = acc2[r] + sbe[l16 + 32] + xs[l16 + 32]; m = m > 0.f ? m : 0.f; atomicAdd(ag + l16 + 32, m);
        m = acc3[r] + sbe[l16 + 48] + xs[l16 + 48]; m = m > 0.f ? m : 0.f; atomicAdd(ag + l16 + 48, m);
    }
}

// ---------------------------------------------------------------------------
// GIN layers 1..3 aggregation: agg[dst] += x[src]   (16 threads per edge)
// ---------------------------------------------------------------------------
__global__ __launch_bounds__(256)
void gin_gather_kernel(const float* __restrict__ x,
                       const int*   __restrict__ ei,
                       float*       __restrict__ agg)
{
    int t   = blockIdx.x * 256 + threadIdx.x;
    int eid = t >> 4;
    int q   = t & 15;
    if (eid >= EE) return;
    int src = ei[eid];
    int dst = ei[EE + eid];
    float4 v = *(const float4*)(x + (size_t)src * DD + q * 4);
    float* a = agg + (size_t)dst * DD + q * 4;
    atomicAdd(a + 0, v.x);
    atomicAdd(a + 1, v.y);
    atomicAdd(a + 2, v.z);
    atomicAdd(a + 3, v.w);
}

// ---------------------------------------------------------------------------
// MLP: h = relu?( relu((x+agg)@W1 folded-BN) @ W2 + b2 )
// 128 thr = 4 waves, 16 nodes/wave, 64 nodes/block, grid = 1250 (exact)
// ---------------------------------------------------------------------------
__global__ __launch_bounds__(128)
void gin_mlp_kernel(const float* __restrict__ xin,
                    const float* __restrict__ agg,
                    const float* __restrict__ W1,
                    const float* __restrict__ b1,
                    const float* __restrict__ gamma,
                    const float* __restrict__ beta,
                    const float* __restrict__ W2,
                    const float* __restrict__ b2,
                    const float* __restrict__ rm,
                    const float* __restrict__ rv,
                    float*       __restrict__ xout,
                    int relu_out)
{
    __shared__ float sW1[64 * 64];
    __shared__ float sW2[64 * 64];
    __shared__ float sT[4][16 * 65];     // per-wave padded intermediate tile
    __shared__ float sScale[64], sShift[64], sB2[64];

    const int tid  = threadIdx.x;
    const int lane = tid & 31;
    const int wave = tid >> 5;
    const int half = lane >> 4;
    const int l16  = lane & 15;

    for (int i = tid; i < 64 * 64; i += 128) { sW1[i] = W1[i]; sW2[i] = W2[i]; }
    if (tid < 64) {
        float sc = gamma[tid] * rsqrtf(rv[tid] + BN_EPS);
        sScale[tid] = sc;
        sShift[tid] = (b1[tid] - rm[tid]) * sc + beta[tid];  // b1 folded into BN shift
        sB2[tid]    = b2[tid];
    }

    const int nbase = (blockIdx.x * 4 + wave) * 16;

    // stage h_in = x + agg
    for (int it = lane; it < 16 * 16; it += 32) {
        int row = it >> 4, c4 = it & 15;
        size_t off = (size_t)(nbase + row) * DD + c4 * 4;
        float4 a = *(const float4*)(xin + off);
        float4 g = *(const float4*)(agg + off);
        float* d = &sT[wave][row * 65 + c4 * 4];
        d[0] = a.x + g.x; d[1] = a.y + g.y; d[2] = a.z + g.z; d[3] = a.w + g.w;
    }
    __syncthreads();

    // ---- GEMM1: (x+agg) @ W1 ----
    v8f acc0 = {}, acc1 = {}, acc2 = {}, acc3 = {};
    {
        const float* Arow = &sT[wave][l16 * 65];
#pragma unroll
        for (int kk = 0; kk < 16; ++kk) {
            const int kc = kk * 4 + half * 2;
            v2f a; a.x = Arow[kc]; a.y = Arow[kc + 1];
            const float* Wr0 = &sW1[kc * 64 + l16];
            const float* Wr1 = Wr0 + 64;
            v2f b0, b1v, b2v, b3v;
            b0.x  = Wr0[0];  b0.y  = Wr1[0];
            b1v.x = Wr0[16]; b1v.y = Wr1[16];
            b2v.x = Wr0[32]; b2v.y = Wr1[32];
            b3v.x = Wr0[48]; b3v.y = Wr1[48];
            acc0 = __builtin_amdgcn_wmma_f32_16x16x4_f32(false, a, false, b0,  (short)0, acc0, false, false);
            acc1 = __builtin_amdgcn_wmma_f32_16x16x4_f32(false, a, false, b1v, (short)0, acc1, false, false);
            acc2 = __builtin_amdgcn_wmma_f32_16x16x4_f32(false, a, false, b2v, (short)0, acc2, false, false);
            acc3 = __builtin_amdgcn_wmma_f32_16x16x4_f32(false, a, false, b3v, (short)0, acc3, false, false);
        }
    }
    __syncthreads();

    // BN(scale,shift) + ReLU, write back to sT (C layout -> memory layout)
#pragma unroll
    for (int r = 0; r < 8; ++r) {
        const int row = r + half * 8;
        float v;
        v = acc0[r] * sScale[l16]      + sShift[l16];      sT[wave][row * 65 + l16]      = v > 0.f ? v : 0.f;
        v = acc1[r] * sScale[l16 + 16] + sShift[l16 + 16]; sT[wave][row * 65 + l16 + 16] = v > 0.f ? v : 0.f;
        v = acc2[r] * sScale[l16 + 32] + sShift[l16 + 32]; sT[wave][row * 65 + l16 + 32] = v > 0.f ? v : 0.f;
        v = acc3[r] * sScale[l16 + 48] + sShift[l16 + 48]; sT[wave][row * 65 + l16 + 48] = v > 0.f ? v : 0.f;
    }
    __syncthreads();

    // ---- GEMM2: t @ W2 ----
    acc0 = (v8f){}; acc1 = (v8f){}; acc2 = (v8f){}; acc3 = (v8f){};
    {
        const float* Arow = &sT[wave][l16 * 65];
#pragma unroll
        for (int kk = 0; kk < 16; ++kk) {
            const int kc = kk * 4 + half * 2;
            v2f a; a.x = Arow[kc]; a.y = Arow[kc + 1];
            const float* Wr0 = &sW2[kc * 64 + l16];
            const float* Wr1 = Wr0 + 64;
            v2f b0, b1v, b2v, b3v;
            b0.x  = Wr0[0];  b0.y  = Wr1[0];
            b1v.x = Wr0[16]; b1v.y = Wr1[16];
            b2v.x = Wr0[32]; b2v.y = Wr1[32];
            b3v.x = Wr0[48]; b3v.y = Wr1[48];
            acc0 = __builtin_amdgcn_wmma_f32_16x16x4_f32(false, a, false, b0,  (short)0, acc0, false, false);
            acc1 = __builtin_amdgcn_wmma_f32_16x16x4_f32(false, a, false, b1v, (short)0, acc1, false, false);
            acc2 = __builtin_amdgcn_wmma_f32_16x16x4_f32(false, a, false, b2v, (short)0, acc2, false, false);
            acc3 = __builtin_amdgcn_wmma_f32_16x16x4_f32(false, a, false, b3v, (short)0, acc3, false, false);
        }
    }

    // bias + optional relu + store
#pragma unroll
    for (int r = 0; r < 8; ++r) {
        const int row = r + half * 8;
        float* o = xout + (size_t)(nbase + row) * DD;
        float v;
        v = acc0[r] + sB2[l16];      if (relu_out) v = v > 0.f ? v : 0.f; o[l16]      = v;
        v = acc1[r] + sB2[l16 + 16]; if (relu_out) v = v > 0.f ? v : 0.f; o[l16 + 16] = v;
        v = acc2[r] + sB2[l16 + 32]; if (relu_out) v = v > 0.f ? v : 0.f; o[l16 + 32] = v;
        v = acc3[r] + sB2[l16 + 48]; if (relu_out) v = v > 0.f ? v : 0.f; o[l16 + 48] = v;
    }
}

// ---------------------------------------------------------------------------
// global mean pool
// ---------------------------------------------------------------------------
__global__ __launch_bounds__(256)
void pool_accum_kernel(const float* __restrict__ x,
                       const int*   __restrict__ batch,
                       float* __restrict__ sums,
                       float* __restrict__ cnts)
{
    int t = blockIdx.x * 256 + threadIdx.x;
    int node = t >> 4;
    int q    = t & 15;
    if (node >= NN) return;
    int g = batch[node];
    float4 v = *(const float4*)(x + (size_t)node * DD + q * 4);
    float* s = sums + (size_t)g * DD + q * 4;
    atomicAdd(s + 0, v.x);
    atomicAdd(s + 1, v.y);
    atomicAdd(s + 2, v.z);
    atomicAdd(s + 3, v.w);
    if (q == 0) atomicAdd(cnts + g, 1.0f);
}

__global__ void pool_final_kernel(const float* __restrict__ sums,
                                  const float* __restrict__ cnts,
                                  float* __restrict__ out)
{
    int i = blockIdx.x * 256 + threadIdx.x;
    if (i >= GG * DD) return;
    float c = cnts[i >> 6];
    out[i] = sums[i] / fmaxf(c, 1.0f);
}

// ---------------------------------------------------------------------------
extern "C" void kernel_launch(void* const* d_in, const int* in_sizes, int n_in,
                              void* d_out, int out_size, void* d_ws, size_t ws_size,
                              hipStream_t stream)
{
    // setup_inputs() dict order (recursive insertion order):
    // x, edge_index, edge_attr, batch,
    // params.layers[i]{W1,b1,gamma,beta,W2,b2} i=0..3, params.We, params.be,
    // stats[i]{rm,rv} i=0..3
    const float* x     = (const float*)d_in[0];
    const int*   ei    = (const int*)  d_in[1];
    const float* ea    = (const float*)d_in[2];
    const int*   batch = (const int*)  d_in[3];

    const float *W1[4], *b1[4], *gamma[4], *beta[4], *W2[4], *b2[4], *rm[4], *rv[4];
    for (int i = 0; i < 4; ++i) {
        W1[i]    = (const float*)d_in[4 + 6 * i + 0];
        b1[i]    = (const float*)d_in[4 + 6 * i + 1];
        gamma[i] = (const float*)d_in[4 + 6 * i + 2];
        beta[i]  = (const float*)d_in[4 + 6 * i + 3];
        W2[i]    = (const float*)d_in[4 + 6 * i + 4];
        b2[i]    = (const float*)d_in[4 + 6 * i + 5];
    }
    const float* We = (const float*)d_in[28];
    const float* be = (const float*)d_in[29];
    for (int i = 0; i < 4; ++i) {
        rm[i] = (const float*)d_in[30 + 2 * i];
        rv[i] = (const float*)d_in[31 + 2 * i];
    }

    // workspace: agg | xa | xb | sums | cnts   (~61.5 MB)
    float* ws   = (float*)d_ws;
    float* agg  = ws;
    float* xa   = agg + (size_t)NN * DD;
    float* xb   = xa  + (size_t)NN * DD;
    float* sums = xb  + (size_t)NN * DD;
    float* cnts = sums + (size_t)GG * DD;

    const int ND = NN * DD;               // 5.12M
    const int zgrid = (ND + 255) / 256;

    // ---- layer 0 (GINE) ----
    gnn_zero_kernel<<<zgrid, 256, 0, stream>>>(agg, ND);
    gine_edge_kernel<<<EE / 128, 256, 0, stream>>>(x, ei, ea, We, be, agg);
    gin_mlp_kernel<<<NN / 64, 128, 0, stream>>>(x, agg, W1[0], b1[0], gamma[0], beta[0],
                                                W2[0], b2[0], rm[0], rv[0], xa, 1);

    // ---- layers 1..3 (GIN) ----
    const float* cur = xa;
    float* nxt = xb;
    for (int L = 1; L < 4; ++L) {
        gnn_zero_kernel<<<zgrid, 256, 0, stream>>>(agg, ND);
        gin_gather_kernel<<<(EE * 16) / 256, 256, 0, stream>>>(cur, ei, agg);
        gin_mlp_kernel<<<NN / 64, 128, 0, stream>>>(cur, agg, W1[L], b1[L], gamma[L], beta[L],
                                                    W2[L], b2[L], rm[L], rv[L], nxt, (L != 3) ? 1 : 0);
        const float* t = cur; cur = nxt; nxt = (float*)t;
    }

    // ---- global mean pool ----
    gnn_zero_kernel<<<(GG * DD + GG + 255) / 256, 256, 0, stream>>>(sums, GG * DD + GG);
    pool_accum_kernel<<<(NN * 16 + 255) / 256, 256, 0, stream>>>(cur, batch, sums, cnts);
    pool_final_kernel<<<(GG * DD + 255) / 256, 256, 0, stream>>>(sums, cnts, (float*)d_out);
}